// MyGATv2_36344013259386
// MI455X (gfx1250) — compile-verified
//
#include <hip/hip_runtime.h>
#include <hip/hip_bf16.h>

typedef __attribute__((ext_vector_type(2))) float v2f;
typedef __attribute__((ext_vector_type(8))) float v8f;

#define NEG_SLOPE 0.2f

__device__ __forceinline__ float lrelu(float v) { return v > 0.f ? v : NEG_SLOPE * v; }

__device__ __forceinline__ void atomicMaxFloat(float* addr, float val) {
    // sign-split trick: integer atomics only; init value must be -inf bits
    if (val >= 0.f) atomicMax((int*)addr, __float_as_int(val));
    else            atomicMin((unsigned int*)addr, __float_as_uint(val));
}

// ---------------------------------------------------------------------------
// init: amax = -inf, denom = 0, accumulators = 0
// ---------------------------------------------------------------------------
__global__ void init_ws_kernel(float* __restrict__ amax0, float* __restrict__ den0,
                               float* __restrict__ amax1, float* __restrict__ den1,
                               float* __restrict__ acc0, float* __restrict__ acc1, int N) {
    int idx = blockIdx.x * blockDim.x + threadIdx.x;
    float ninf = -__builtin_inff();
    if (idx < N * 8) { amax0[idx] = ninf; den0[idx] = 0.f; }
    if (idx < N)     { amax1[idx] = ninf; den1[idx] = 0.f; }
    if (idx < N * 64) acc0[idx] = 0.f;
    if (idx < N * 16) acc1[idx] = 0.f;
}

// ---------------------------------------------------------------------------
// Layer-0 GEMM: tx0[N x 128] = X[N x 128] * [Wsrc^T | Wdst^T]
// (cols 0-63 = src transform, 64-127 = dst transform; one buffer, no
//  divergent store selection). 8 waves/block, wave w = column tile w.
// K=128 -> 32 x V_WMMA_F32_16X16X4_F32 per wave.
// ---------------------------------------------------------------------------
__global__ void __launch_bounds__(256) gemm_tx0_kernel(
        const float* __restrict__ X, const float* __restrict__ Wsrc,
        const float* __restrict__ Wdst, float* __restrict__ tx, int N) {
    __shared__ float xs[16 * 128];
    const int rowbase = blockIdx.x * 16;
    const int t = threadIdx.x;

    // stage 16x128 tile of X (rows past N -> 0)
    #pragma unroll
    for (int it = 0; it < 2; ++it) {
        int idx = t + it * 256;          // float4 index 0..511
        int r  = idx >> 5;               // 32 float4 per row
        int cc = (idx & 31) * 4;
        int row = rowbase + r;
        float4 v = make_float4(0.f, 0.f, 0.f, 0.f);
        if (row < N) v = *(const float4*)(X + (size_t)row * 128 + cc);
        *(float4*)(xs + r * 128 + cc) = v;
    }
    __syncthreads();

    const int wave = t >> 5;             // 0..7 = column tile
    const int lane = t & 31;
    const int lr   = lane & 15;
    const int koff = (lane >> 4) * 2;    // lanes 16-31 hold K+2,K+3
    const int col  = wave * 16 + lr;     // 0..127
    const float* Wrow = (col < 64) ? (Wsrc + (size_t)col * 128)
                                   : (Wdst + (size_t)(col - 64) * 128);

    v8f acc = {};
    #pragma unroll 4
    for (int kb = 0; kb < 128; kb += 4) {
        float2 av = *(const float2*)(xs + lr * 128 + kb + koff);
        float2 bv = *(const float2*)(Wrow + kb + koff);
        v2f a; a.x = av.x; a.y = av.y;
        v2f b; b.x = bv.x; b.y = bv.y;
        acc = __builtin_amdgcn_wmma_f32_16x16x4_f32(
                false, a, false, b, (short)0, acc, false, false);
    }

    const int rq = (lane >> 4) * 8;      // D: M = v + (lane>=16 ? 8 : 0)
    if (rowbase + 16 <= N) {             // full tile: unconditional stores
        #pragma unroll
        for (int v = 0; v < 8; ++v)
            tx[(size_t)(rowbase + v + rq) * 128 + col] = acc[v];
    } else {
        #pragma unroll
        for (int v = 0; v < 8; ++v) {
            int row = rowbase + v + rq;
            if (row < N) tx[(size_t)row * 128 + col] = acc[v];
        }
    }
}

// ---------------------------------------------------------------------------
// Layer-1 GEMM: tx1[N x 32] = H[N x 64] * [Wsrc1^T | Wdst1^T]
// 2 waves/block. K=64 -> 16 WMMA.
// ---------------------------------------------------------------------------
__global__ void __launch_bounds__(64) gemm_tx1_kernel(
        const float* __restrict__ H, const float* __restrict__ Wsrc,
        const float* __restrict__ Wdst, float* __restrict__ tx, int N) {
    __shared__ float xs[16 * 64];
    const int rowbase = blockIdx.x * 16;
    const int t = threadIdx.x;

    #pragma unroll
    for (int it = 0; it < 4; ++it) {
        int idx = t + it * 64;           // float4 index 0..255
        int r  = idx >> 4;               // 16 float4 per row
        int cc = (idx & 15) * 4;
        int row = rowbase + r;
        float4 v = make_float4(0.f, 0.f, 0.f, 0.f);
        if (row < N) v = *(const float4*)(H + (size_t)row * 64 + cc);
        *(float4*)(xs + r * 64 + cc) = v;
    }
    __syncthreads();

    const int wave = t >> 5;             // 0..1
    const int lane = t & 31;
    const int lr   = lane & 15;
    const int koff = (lane >> 4) * 2;
    const int col  = wave * 16 + lr;     // 0..31
    const float* Wrow = (col < 16) ? (Wsrc + (size_t)col * 64)
                                   : (Wdst + (size_t)(col - 16) * 64);

    v8f acc = {};
    #pragma unroll 4
    for (int kb = 0; kb < 64; kb += 4) {
        float2 av = *(const float2*)(xs + lr * 64 + kb + koff);
        float2 bv = *(const float2*)(Wrow + kb + koff);
        v2f a; a.x = av.x; a.y = av.y;
        v2f b; b.x = bv.x; b.y = bv.y;
        acc = __builtin_amdgcn_wmma_f32_16x16x4_f32(
                false, a, false, b, (short)0, acc, false, false);
    }

    const int rq = (lane >> 4) * 8;
    if (rowbase + 16 <= N) {
        #pragma unroll
        for (int v = 0; v < 8; ++v)
            tx[(size_t)(rowbase + v + rq) * 32 + col] = acc[v];
    } else {
        #pragma unroll
        for (int v = 0; v < 8; ++v) {
            int row = rowbase + v + rq;
            if (row < N) tx[(size_t)row * 32 + col] = acc[v];
        }
    }
}

// ---------------------------------------------------------------------------
// Layer-0 edge pass A: attention scores -> segment max (8 heads x 8 ch)
// Optionally spills per-edge scores so pass B skips the txd gather.
// ---------------------------------------------------------------------------
__global__ void edge_max0_kernel(const int* __restrict__ src, const int* __restrict__ dst,
                                 int E, int Etot, const float* __restrict__ tx,
                                 const float* __restrict__ attn, float* __restrict__ amax,
                                 float* __restrict__ scores) {
    int e = blockIdx.x * blockDim.x + threadIdx.x;
    if (e >= Etot) return;
    int j, i;
    if (e < E) { j = src[e]; i = dst[e]; } else { j = i = e - E; }
    const float4* sj = (const float4*)(tx + (size_t)j * 128);        // src cols 0-63
    const float4* di = (const float4*)(tx + (size_t)i * 128 + 64);   // dst cols 64-127
    #pragma unroll
    for (int h = 0; h < 8; ++h) {
        float4 s0 = sj[h * 2], s1 = sj[h * 2 + 1];
        float4 d0 = di[h * 2], d1 = di[h * 2 + 1];
        const float* ah = attn + h * 8;
        float sc = lrelu(s0.x + d0.x) * ah[0] + lrelu(s0.y + d0.y) * ah[1]
                 + lrelu(s0.z + d0.z) * ah[2] + lrelu(s0.w + d0.w) * ah[3]
                 + lrelu(s1.x + d1.x) * ah[4] + lrelu(s1.y + d1.y) * ah[5]
                 + lrelu(s1.z + d1.z) * ah[6] + lrelu(s1.w + d1.w) * ah[7];
        if (scores) scores[(size_t)e * 8 + h] = sc;
        atomicMaxFloat(amax + (size_t)i * 8 + h, sc);
    }
}

// ---------------------------------------------------------------------------
// Layer-0 edge pass B: w = exp(score - max); denom += w; acc += txs[j] * w
// ---------------------------------------------------------------------------
__global__ void edge_agg0_kernel(const int* __restrict__ src, const int* __restrict__ dst,
                                 int E, int Etot, const float* __restrict__ tx,
                                 const float* __restrict__ attn, const float* __restrict__ amax,
                                 float* __restrict__ den, float* __restrict__ acc,
                                 const float* __restrict__ scores) {
    int e = blockIdx.x * blockDim.x + threadIdx.x;
    if (e >= Etot) return;
    int j, i;
    if (e < E) { j = src[e]; i = dst[e]; } else { j = i = e - E; }
    const float4* sj = (const float4*)(tx + (size_t)j * 128);
    const float4* di = (const float4*)(tx + (size_t)i * 128 + 64);
    #pragma unroll
    for (int h = 0; h < 8; ++h) {
        float4 s0 = sj[h * 2], s1 = sj[h * 2 + 1];
        float sc;
        if (scores) {
            sc = scores[(size_t)e * 8 + h];
        } else {
            float4 d0 = di[h * 2], d1 = di[h * 2 + 1];
            const float* ah = attn + h * 8;
            sc = lrelu(s0.x + d0.x) * ah[0] + lrelu(s0.y + d0.y) * ah[1]
               + lrelu(s0.z + d0.z) * ah[2] + lrelu(s0.w + d0.w) * ah[3]
               + lrelu(s1.x + d1.x) * ah[4] + lrelu(s1.y + d1.y) * ah[5]
               + lrelu(s1.z + d1.z) * ah[6] + lrelu(s1.w + d1.w) * ah[7];
        }
        float w = __expf(sc - amax[(size_t)i * 8 + h]);
        atomicAdd(den + (size_t)i * 8 + h, w);
        float* ai = acc + (size_t)i * 64 + h * 8;
        atomicAdd(ai + 0, s0.x * w); atomicAdd(ai + 1, s0.y * w);
        atomicAdd(ai + 2, s0.z * w); atomicAdd(ai + 3, s0.w * w);
        atomicAdd(ai + 4, s1.x * w); atomicAdd(ai + 5, s1.y * w);
        atomicAdd(ai + 6, s1.z * w); atomicAdd(ai + 7, s1.w * w);
    }
}

// ---------------------------------------------------------------------------
// Layer-0 epilogue: h = elu(acc/den + bias)  (writes H[N x 64])
// ---------------------------------------------------------------------------
__global__ void epilogue0_kernel(const float* __restrict__ acc, const float* __restrict__ den,
                                 const float* __restrict__ bias, float* __restrict__ hout, int N) {
    int idx = blockIdx.x * blockDim.x + threadIdx.x;
    if (idx >= N * 64) return;
    int i = idx >> 6;
    int k = idx & 63;
    float v = acc[idx] / den[(size_t)i * 8 + (k >> 3)] + bias[k];
    hout[idx] = v > 0.f ? v : (__expf(v) - 1.f);
}

// ---------------------------------------------------------------------------
// Layer-1 edge pass A (1 head, 16 channels)
// ---------------------------------------------------------------------------
__global__ void edge_max1_kernel(const int* __restrict__ src, const int* __restrict__ dst,
                                 int E, int Etot, const float* __restrict__ tx,
                                 const float* __restrict__ attn, float* __restrict__ amax,
                                 float* __restrict__ scores) {
    int e = blockIdx.x * blockDim.x + threadIdx.x;
    if (e >= Etot) return;
    int j, i;
    if (e < E) { j = src[e]; i = dst[e]; } else { j = i = e - E; }
    const float4* sj = (const float4*)(tx + (size_t)j * 32);        // src cols 0-15
    const float4* di = (const float4*)(tx + (size_t)i * 32 + 16);   // dst cols 16-31
    float sc = 0.f;
    #pragma unroll
    for (int q = 0; q < 4; ++q) {
        float4 s = sj[q], d = di[q];
        const float* aq = attn + q * 4;
        sc += lrelu(s.x + d.x) * aq[0] + lrelu(s.y + d.y) * aq[1]
            + lrelu(s.z + d.z) * aq[2] + lrelu(s.w + d.w) * aq[3];
    }
    if (scores) scores[e] = sc;
    atomicMaxFloat(amax + i, sc);
}

// ---------------------------------------------------------------------------
// Layer-1 edge pass B
// ---------------------------------------------------------------------------
__global__ void edge_agg1_kernel(const int* __restrict__ src, const int* __restrict__ dst,
                                 int E, int Etot, const float* __restrict__ tx,
                                 const float* __restrict__ attn, const float* __restrict__ amax,
                                 float* __restrict__ den, float* __restrict__ acc,
                                 const float* __restrict__ scores) {
    int e = blockIdx.x * blockDim.x + threadIdx.x;
    if (e >= Etot) return;
    int j, i;
    if (e < E) { j = src[e]; i = dst[e]; } else { j = i = e - E; }
    const float4* sj = (const float4*)(tx + (size_t)j * 32);
    const float4* di = (const float4*)(tx + (size_t)i * 32 + 16);
    float4 S[4];
    float sc = 0.f;
    #pragma unroll
    for (int q = 0; q < 4; ++q) S[q] = sj[q];
    if (scores) {
        sc = scores[e];
    } else {
        #pragma unroll
        for (int q = 0; q < 4; ++q) {
            float4 s = S[q], d = di[q];
            const float* aq = attn + q * 4;
            sc += lrelu(s.x + d.x) * aq[0] + lrelu(s.y + d.y) * aq[1]
                + lrelu(s.z + d.z) * aq[2] + lrelu(s.w + d.w) * aq[3];
        }
    }
    float w = __expf(sc - amax[i]);
    atomicAdd(den + i, w);
    float* ai = acc + (size_t)i * 16;
    #pragma unroll
    for (int q = 0; q < 4; ++q) {
        atomicAdd(ai + q * 4 + 0, S[q].x * w);
        atomicAdd(ai + q * 4 + 1, S[q].y * w);
        atomicAdd(ai + q * 4 + 2, S[q].z * w);
        atomicAdd(ai + q * 4 + 3, S[q].w * w);
    }
}

// ---------------------------------------------------------------------------
// Layer-1 epilogue: softmax(acc/den + bias) per node -> d_out[N x 16]
// ---------------------------------------------------------------------------
__global__ void epilogue1_kernel(const float* __restrict__ acc, const float* __restrict__ den,
                                 const float* __restrict__ bias, float* __restrict__ out, int N) {
    int i = blockIdx.x * blockDim.x + threadIdx.x;
    if (i >= N) return;
    float invd = 1.f / den[i];
    float t[16];
    float m = -__builtin_inff();
    #pragma unroll
    for (int c = 0; c < 16; ++c) {
        t[c] = acc[(size_t)i * 16 + c] * invd + bias[c];
        m = fmaxf(m, t[c]);
    }
    float s = 0.f;
    #pragma unroll
    for (int c = 0; c < 16; ++c) { t[c] = __expf(t[c] - m); s += t[c]; }
    float inv = 1.f / s;
    #pragma unroll
    for (int c = 0; c < 16; ++c) out[(size_t)i * 16 + c] = t[c] * inv;
}

// ---------------------------------------------------------------------------
// Host launcher
// ---------------------------------------------------------------------------
extern "C" void kernel_launch(void* const* d_in, const int* in_sizes, int n_in,
                              void* d_out, int out_size, void* d_ws, size_t ws_size,
                              hipStream_t stream) {
    const float* x     = (const float*)d_in[0];
    const int*   edges = (const int*)d_in[1];
    const float* Wsrc0 = (const float*)d_in[2];
    const float* Wdst0 = (const float*)d_in[3];
    const float* attn0 = (const float*)d_in[4];
    const float* bias0 = (const float*)d_in[5];
    const float* Wsrc1 = (const float*)d_in[6];
    const float* Wdst1 = (const float*)d_in[7];
    const float* attn1 = (const float*)d_in[8];
    const float* bias1 = (const float*)d_in[9];
    float* out = (float*)d_out;

    const int N = in_sizes[0] / 128;
    const int E = in_sizes[1] / 2;
    const int Etot = E + N;               // with self loops
    const int* src = edges;
    const int* dst = edges + E;

    // workspace layout (floats)
    float* ws = (float*)d_ws;
    size_t off = 0;
    float* tx0   = ws + off; off += (size_t)N * 128;  // [src|dst] transforms, layer 0
    float* acc0  = ws + off; off += (size_t)N * 64;
    float* amax0 = ws + off; off += (size_t)N * 8;
    float* den0  = ws + off; off += (size_t)N * 8;
    float* amax1 = ws + off; off += (size_t)N;
    float* den1  = ws + off; off += (size_t)N;
    float* acc1  = ws + off; off += (size_t)N * 16;
    size_t base_floats = off;

    // optional per-edge score spill (saves the txd gather in pass B)
    size_t score_floats = (size_t)Etot * 9;
    bool use_scores = ws_size >= (base_floats + score_floats) * sizeof(float);
    float* scores0 = use_scores ? (ws + base_floats) : nullptr;
    float* scores1 = use_scores ? (ws + base_floats + (size_t)Etot * 8) : nullptr;

    // reuse (stream-ordered, no hazards):
    float* h0  = ws;                      // overwrites tx0 head after pass B0 done
    float* tx1 = acc0;                    // overwrites acc0 after epilogue0 done

    const int B = 256;
    init_ws_kernel<<<(N * 64 + B - 1) / B, B, 0, stream>>>(amax0, den0, amax1, den1, acc0, acc1, N);
    // layer 0
    gemm_tx0_kernel<<<(N + 15) / 16, 256, 0, stream>>>(x, Wsrc0, Wdst0, tx0, N);
    edge_max0_kernel<<<(Etot + B - 1) / B, B, 0, stream>>>(src, dst, E, Etot, tx0, attn0, amax0, scores0);
    edge_agg0_kernel<<<(Etot + B - 1) / B, B, 0, stream>>>(src, dst, E, Etot, tx0, attn0, amax0, den0, acc0, scores0);
    epilogue0_kernel<<<(N * 64 + B - 1) / B, B, 0, stream>>>(acc0, den0, bias0, h0, N);
    // layer 1
    gemm_tx1_kernel<<<(N + 15) / 16, 64, 0, stream>>>(h0, Wsrc1, Wdst1, tx1, N);
    edge_max1_kernel<<<(Etot + B - 1) / B, B, 0, stream>>>(src, dst, E, Etot, tx1, attn1, amax1, scores1);
    edge_agg1_kernel<<<(Etot + B - 1) / B, B, 0, stream>>>(src, dst, E, Etot, tx1, attn1, amax1, den1, acc1, scores1);
    epilogue1_kernel<<<(N + B - 1) / B, B, 0, stream>>>(acc1, den1, bias1, out, N);
}